// SPR_RGCN_88648124990153
// MI455X (gfx1250) — compile-verified
//
#include <hip/hip_runtime.h>

#define NUM_REL   8
#define NUM_GRAPHS 64
#define HID       128
#define OUT_DIM   16
#define N_NODES   65536
#define N_EDGES   1048576
#define SB_STRIDE 68   // LDS row stride (floats) for transposed weight chunk

typedef __attribute__((ext_vector_type(2))) float v2f;
typedef __attribute__((ext_vector_type(8))) float v8f;

// ---------------------------------------------------------------------------
// Per-(dst, relation) in-degree counts (graph is static -> computed once).
__global__ __launch_bounds__(256) void rgcn_count_kernel(
    const int* __restrict__ dst, const int* __restrict__ et,
    int* __restrict__ cnt) {
  int e = blockIdx.x * 256 + threadIdx.x;
  if (e < N_EDGES) atomicAdd(&cnt[dst[e] * NUM_REL + et[e]], 1);
}

// ---------------------------------------------------------------------------
// acc[dst][rel][:] += x[src][:]   (one wave32 per edge, float4 per lane)
__global__ __launch_bounds__(256) void rgcn_scatter_kernel(
    const float* __restrict__ x, const int* __restrict__ src,
    const int* __restrict__ dst, const int* __restrict__ et,
    float* __restrict__ acc) {
  int edge = blockIdx.x * 8 + (threadIdx.x >> 5);
  int lane = threadIdx.x & 31;
  if (edge >= N_EDGES) return;
  int s = src[edge], d = dst[edge], r = et[edge];
  const float4 v = *(const float4*)(x + (size_t)s * HID + lane * 4);
  float* p = acc + ((size_t)d * NUM_REL + r) * HID + lane * 4;
  unsafeAtomicAdd(p + 0, v.x);
  unsafeAtomicAdd(p + 1, v.y);
  unsafeAtomicAdd(p + 2, v.z);
  unsafeAtomicAdd(p + 3, v.w);
}

// ---------------------------------------------------------------------------
// out[i,:] = relu( sum_r (acc[i,r,:]*norm(i,r)) @ W_r + x[i,:] @ root + b )
// 8 waves/block, 16 rows/wave -> 128 rows/block, 128 cols, K = 9*128.
// Weight chunks (64 K-rows) staged TRANSPOSED in LDS: sB[n][k], stride 68,
// so each B fragment (W[k][n], W[k+1][n]) is one conflict-free ds_load_b64
// landing directly in a WMMA-ready even VGPR pair.
__global__ __launch_bounds__(256) void rgcn_gemm_kernel(
    const float* __restrict__ acc, const int* __restrict__ cnt,
    const float* __restrict__ xin, const float* __restrict__ W,
    const float* __restrict__ root, const float* __restrict__ bias,
    float* __restrict__ out) {
  __shared__ float sB[HID * SB_STRIDE];  // 34.8 KB

  const int lane = threadIdx.x & 31;
  const int wave = threadIdx.x >> 5;
  const int mr   = lane & 15;          // row within 16-row tile (== N index)
  const int kh   = (lane >> 4) << 1;   // K sub-offset: 0 (lanes 0-15) / 2 (16-31)
  const int row  = blockIdx.x * 128 + wave * 16 + mr;

  v8f c[8];
#pragma unroll
  for (int t = 0; t < 8; ++t) {
    float bv = bias[t * 16 + mr];
    c[t] = (v8f){bv, bv, bv, bv, bv, bv, bv, bv};
  }

  for (int s = 0; s < NUM_REL + 1; ++s) {
    const float* Ws;
    const float* As;
    float norm;
    if (s < NUM_REL) {
      Ws = W + (size_t)s * HID * HID;
      As = acc + ((size_t)row * NUM_REL + s) * HID;
      int cc = cnt[row * NUM_REL + s];
      norm = 1.0f / (float)(cc > 1 ? cc : 1);
    } else {
      Ws = root;
      As = xin + (size_t)row * HID;
      norm = 1.0f;
    }

    for (int k0 = 0; k0 < HID; k0 += 64) {
      __syncthreads();
      // Stage 64x128 chunk transposed: sB[n*68 + kk] = Ws[(k0+kk)*HID + n].
      // Consecutive lanes -> consecutive n: all 4 global b32 loads coalesced;
      // LDS write is one 16B-aligned ds_store_b128 per thread per item.
      for (int i = threadIdx.x; i < (64 / 4) * HID; i += 256) {
        int n  = i & (HID - 1);
        int k4 = (i >> 7) << 2;
        float4 w;
        w.x = Ws[(size_t)(k0 + k4 + 0) * HID + n];
        w.y = Ws[(size_t)(k0 + k4 + 1) * HID + n];
        w.z = Ws[(size_t)(k0 + k4 + 2) * HID + n];
        w.w = Ws[(size_t)(k0 + k4 + 3) * HID + n];
        *(float4*)(sB + n * SB_STRIDE + k4) = w;
      }
      __syncthreads();

#pragma unroll
      for (int kk = 0; kk < 64; kk += 4) {
        v2f a = *(const v2f*)(As + k0 + kk + kh);  // A 16x4: K = kh, kh+1
        a *= norm;                                 // fold mean normalization
#pragma unroll
        for (int t = 0; t < 8; ++t) {
          v2f b = *(const v2f*)(sB + (t * 16 + mr) * SB_STRIDE + kk + kh);
          c[t] = __builtin_amdgcn_wmma_f32_16x16x4_f32(
              false, a, false, b, (short)0, c[t], false, false);
        }
      }
    }
  }

  // relu + store: C/D element v -> M = v + (lane>=16 ? 8 : 0), N = lane&15
  const int mbase = blockIdx.x * 128 + wave * 16 + ((lane >> 4) << 3);
#pragma unroll
  for (int t = 0; t < 8; ++t) {
#pragma unroll
    for (int v = 0; v < 8; ++v) {
      float val = c[t][v];
      val = val > 0.0f ? val : 0.0f;
      out[(size_t)(mbase + v) * HID + t * 16 + mr] = val;
    }
  }
}

// ---------------------------------------------------------------------------
// Global mean pool accumulate: one wave per node.
__global__ __launch_bounds__(256) void rgcn_pool_kernel(
    const float* __restrict__ h, const int* __restrict__ batch,
    float* __restrict__ pool, int* __restrict__ pcnt) {
  int node = blockIdx.x * 8 + (threadIdx.x >> 5);
  int lane = threadIdx.x & 31;
  if (node >= N_NODES) return;
  int g = batch[node];
  const float4 v = *(const float4*)(h + (size_t)node * HID + lane * 4);
  float* p = pool + g * HID + lane * 4;
  unsafeAtomicAdd(p + 0, v.x);
  unsafeAtomicAdd(p + 1, v.y);
  unsafeAtomicAdd(p + 2, v.z);
  unsafeAtomicAdd(p + 3, v.w);
  if (lane == 0) atomicAdd(&pcnt[g], 1);
}

// ---------------------------------------------------------------------------
// out[g][o] = (pool[g]/max(cnt,1)) @ lin_w + lin_b   (64x16, trivial)
__global__ __launch_bounds__(256) void rgcn_final_kernel(
    const float* __restrict__ pool, const int* __restrict__ pcnt,
    const float* __restrict__ lw, const float* __restrict__ lb,
    float* __restrict__ out) {
  int t = blockIdx.x * 256 + threadIdx.x;
  if (t >= NUM_GRAPHS * OUT_DIM) return;
  int g = t >> 4, o = t & 15;
  int cc = pcnt[g];
  float inv = 1.0f / (float)(cc > 1 ? cc : 1);
  float s = lb[o];
  for (int c = 0; c < HID; ++c)
    s += pool[g * HID + c] * inv * lw[c * OUT_DIM + o];
  out[t] = s;
}

// ---------------------------------------------------------------------------
extern "C" void kernel_launch(void* const* d_in, const int* in_sizes, int n_in,
                              void* d_out, int out_size, void* d_ws, size_t ws_size,
                              hipStream_t stream) {
  const float* x     = (const float*)d_in[0];
  const int*   eidx  = (const int*)d_in[1];
  const int*   et    = (const int*)d_in[2];
  const int*   batch = (const int*)d_in[3];
  const float* W1    = (const float*)d_in[4];
  const float* root1 = (const float*)d_in[5];
  const float* b1    = (const float*)d_in[6];
  const float* W2    = (const float*)d_in[7];
  const float* root2 = (const float*)d_in[8];
  const float* b2    = (const float*)d_in[9];
  const float* lw    = (const float*)d_in[10];
  const float* lb    = (const float*)d_in[11];
  const int* src = eidx;
  const int* dst = eidx + N_EDGES;

  char* ws = (char*)d_ws;
  float* acc  = (float*)ws; ws += (size_t)N_NODES * NUM_REL * HID * 4;  // 256 MB
  float* h1   = (float*)ws; ws += (size_t)N_NODES * HID * 4;            // 32 MB
  float* h2   = (float*)ws; ws += (size_t)N_NODES * HID * 4;            // 32 MB
  int*   cnt  = (int*)ws;   ws += (size_t)N_NODES * NUM_REL * 4;        // 2 MB
  float* pool = (float*)ws; ws += NUM_GRAPHS * HID * 4;
  int*   pcnt = (int*)ws;   ws += NUM_GRAPHS * 4;

  // per-(dst,rel) degree: graph identical for both layers -> compute once
  hipMemsetAsync(cnt, 0, (size_t)N_NODES * NUM_REL * 4, stream);
  rgcn_count_kernel<<<(N_EDGES + 255) / 256, 256, 0, stream>>>(dst, et, cnt);

  // layer 1
  hipMemsetAsync(acc, 0, (size_t)N_NODES * NUM_REL * HID * 4, stream);
  rgcn_scatter_kernel<<<N_EDGES / 8, 256, 0, stream>>>(x, src, dst, et, acc);
  rgcn_gemm_kernel<<<N_NODES / 128, 256, 0, stream>>>(acc, cnt, x, W1, root1, b1, h1);

  // layer 2
  hipMemsetAsync(acc, 0, (size_t)N_NODES * NUM_REL * HID * 4, stream);
  rgcn_scatter_kernel<<<N_EDGES / 8, 256, 0, stream>>>(h1, src, dst, et, acc);
  rgcn_gemm_kernel<<<N_NODES / 128, 256, 0, stream>>>(acc, cnt, h1, W2, root2, b2, h2);

  // global mean pool + output linear
  hipMemsetAsync(pool, 0, (size_t)(NUM_GRAPHS * HID + NUM_GRAPHS) * 4, stream);
  rgcn_pool_kernel<<<N_NODES / 8, 256, 0, stream>>>(h2, batch, pool, pcnt);
  rgcn_final_kernel<<<(NUM_GRAPHS * OUT_DIM + 255) / 256, 256, 0, stream>>>(
      pool, pcnt, lw, lb, (float*)d_out);
}